// GCNModelRNN_sparse_6743098655056
// MI455X (gfx1250) — compile-verified
//
#include <hip/hip_runtime.h>
#include <cstdint>
#include <cstddef>

// Problem constants (match reference).
#define N_NODES 50000
#define N_EDGES 800000
#define T_STEPS 50
#define HID     64
#define KDIM    65   // 1 + H  (== H + 1)
#define KPAD    68   // pad K to multiple of 4 for V_WMMA_F32_16X16X4_F32
#define RUW     128  // gate output width, layer 1

typedef __attribute__((ext_vector_type(2))) float v2f;
typedef __attribute__((ext_vector_type(8))) float v8f;

__device__ __forceinline__ float sigmoidf_(float x) {
  return 1.0f / (1.0f + __expf(-x));
}

// ---------------------------------------------------------------- utilities
__global__ void fill_f32(float* p, float v, int n) {
  int i = blockIdx.x * 256 + threadIdx.x;
  if (i < n) p[i] = v;
}
__global__ void fill_i32(int* p, int v, int n) {
  int i = blockIdx.x * 256 + threadIdx.x;
  if (i < n) p[i] = v;
}
__global__ void copy_i32(const int* __restrict__ a, int* __restrict__ b, int n) {
  int i = blockIdx.x * 256 + threadIdx.x;
  if (i < n) b[i] = a[i];
}

// ------------------------------------------------------------- CSR building
__global__ void hist_rows(const int* __restrict__ idx, int* __restrict__ deg) {
  int e = blockIdx.x * 256 + threadIdx.x;
  if (e < N_EDGES) atomicAdd(&deg[idx[2 * e]], 1);
}

// Single-workgroup chunked Hillis–Steele scan: rowptr = exclusive_scan(deg).
__global__ __launch_bounds__(1024) void scan_deg(const int* __restrict__ deg,
                                                 int* __restrict__ rowptr, int n) {
  __shared__ int sh[1024];
  __shared__ int carry;
  if (threadIdx.x == 0) { carry = 0; rowptr[0] = 0; }
  __syncthreads();
  for (int base = 0; base < n; base += 1024) {
    int i = base + (int)threadIdx.x;
    int v = (i < n) ? deg[i] : 0;
    sh[threadIdx.x] = v;
    __syncthreads();
    for (int off = 1; off < 1024; off <<= 1) {
      int t = (threadIdx.x >= (unsigned)off) ? sh[threadIdx.x - off] : 0;
      __syncthreads();
      sh[threadIdx.x] += t;
      __syncthreads();
    }
    if (i < n) rowptr[i + 1] = carry + sh[threadIdx.x];  // inclusive -> rowptr[i+1]
    __syncthreads();
    if (threadIdx.x == 1023) carry += sh[1023];
    __syncthreads();
  }
}

__global__ void scatter_csr(const int* __restrict__ idx, const float* __restrict__ val,
                            int* __restrict__ cursor, int* __restrict__ ccol,
                            float* __restrict__ cval) {
  int e = blockIdx.x * 256 + threadIdx.x;
  if (e >= N_EDGES) return;
  int r = idx[2 * e];
  int c = idx[2 * e + 1];
  int p = atomicAdd(&cursor[r], 1);
  ccol[p] = c;
  cval[p] = val[e];
}

// Pad (65, ncols) weight to (68, ncols) with zero rows; layouts share ncols so
// flat indices align for k < 65.
__global__ void pad_weight(const float* __restrict__ W, float* __restrict__ Wp, int ncols) {
  int i = blockIdx.x * 256 + threadIdx.x;
  int total = KPAD * ncols;
  if (i >= total) return;
  Wp[i] = (i < KDIM * ncols) ? W[i] : 0.0f;
}

// --------------------------------------------------------------- xc builders
// Layer 1: xc[n] = [ x_t[n], (use_r ? r[n,:]*h1[n,:] : h1[n,:]), 0,0,0 ]
__global__ void build_xc_l1(const float* __restrict__ xt, const float* __restrict__ h1,
                            const float* __restrict__ RU, int use_r,
                            float* __restrict__ xc) {
  int i = blockIdx.x * 256 + threadIdx.x;
  if (i >= N_NODES * KPAD) return;
  int n = i / KPAD;
  int c = i - n * KPAD;
  float v = 0.0f;
  if (c == 0) {
    v = xt[n];
  } else if (c <= HID) {
    int j = c - 1;
    float h = h1[n * HID + j];
    if (use_r) h *= RU[n * RUW + j];  // r = RU[:, :64]
    v = h;
  }
  xc[i] = v;
}

// Layer 2: xc[n] = [ h1[n,:], (use_r ? r2[n]*h2[n] : h2[n]), 0,0,0 ]
__global__ void build_xc_l2(const float* __restrict__ h1, const float* __restrict__ h2,
                            const float* __restrict__ r2, int use_r,
                            float* __restrict__ xc) {
  int i = blockIdx.x * 256 + threadIdx.x;
  if (i >= N_NODES * KPAD) return;
  int n = i / KPAD;
  int c = i - n * KPAD;
  float v = 0.0f;
  if (c < HID) {
    v = h1[n * HID + c];
  } else if (c == HID) {
    v = h2[n];
    if (use_r) v *= r2[n];
  }
  xc[i] = v;
}

// -------------------------------------------------------------------- SPMM
// CSR SPMM: Y[row,:] = sum_j val[j] * X[col[j],:]   (width KPAD=68)
// One wave per row; lanes cover columns {lane, lane+32, lane+64(<4)}.
__global__ __launch_bounds__(256) void spmm_csr(const int* __restrict__ rowptr,
                                                const int* __restrict__ cols,
                                                const float* __restrict__ vals,
                                                const float* __restrict__ X,
                                                float* __restrict__ Y) {
  const int lane = threadIdx.x & 31;
  const int row = blockIdx.x * 8 + (threadIdx.x >> 5);
  if (row >= N_NODES) return;
  const int beg = rowptr[row];
  const int end = rowptr[row + 1];
  float a0 = 0.f, a1 = 0.f, a2 = 0.f;
  for (int j = beg; j < end; ++j) {
    const int c = cols[j];
    const float v = vals[j];
    const float* xr = X + (size_t)c * KPAD;
    if (j + 1 < end)  // pull next neighbor row toward L0/L2 (global_prefetch_b8)
      __builtin_prefetch(X + (size_t)cols[j + 1] * KPAD, 0, 3);
    a0 += v * xr[lane];
    a1 += v * xr[lane + 32];
    if (lane < 4) a2 += v * xr[lane + 64];
  }
  float* yr = Y + (size_t)row * KPAD;
  yr[lane] = a0;
  yr[lane + 32] = a1;
  if (lane < 4) yr[lane + 64] = a2;
}

// ------------------------------------------------------------- WMMA GEMMs
// Gate: RU = sigmoid(S @ Wp + b), S:(N,68), Wp:(68,128), RU:(N,128)
// One wave computes one 16x16 tile via 17 chained V_WMMA_F32_16X16X4_F32.
__global__ __launch_bounds__(128) void gemm_gate_wmma(const float* __restrict__ S,
                                                      const float* __restrict__ Wp,
                                                      const float* __restrict__ bias,
                                                      float* __restrict__ RU) {
  const int lane = threadIdx.x;            // 0..31
  const int wave = threadIdx.y;            // 0..3
  const int m0 = (blockIdx.x * 4 + wave) * 16;
  if (m0 >= N_NODES) return;               // wave-uniform; EXEC stays all-ones
  const int n0 = blockIdx.y * 16;
  const int l15 = lane & 15;
  const int khi = (lane >> 4) << 1;        // lanes 0-15 -> K 0/1, lanes 16-31 -> K 2/3
  const float* Srow = S + (size_t)(m0 + l15) * KPAD;
  v8f acc = {};
  for (int k = 0; k < KPAD; k += 4) {
    v2f a, b;
    a.x = Srow[k + khi];
    a.y = Srow[k + khi + 1];
    b.x = Wp[(size_t)(k + khi) * RUW + n0 + l15];
    b.y = Wp[(size_t)(k + khi + 1) * RUW + n0 + l15];
    acc = __builtin_amdgcn_wmma_f32_16x16x4_f32(false, a, false, b, (short)0, acc,
                                                false, false);
  }
  const float bv = bias[n0 + l15];
  const int rbase = m0 + ((lane >> 4) << 3);  // +8 for lanes 16-31
#pragma unroll
  for (int j = 0; j < 8; ++j) {
    float v = sigmoidf_(acc[j] + bv);
    RU[(size_t)(rbase + j) * RUW + n0 + l15] = v;
  }
}

// Candidate + GRU update (layer 1):
//   c  = tanh(S @ Wp + b)         (N,64)
//   h1 = u*h1 + (1-u)*c,  u = RU[:, 64:128]
__global__ __launch_bounds__(128) void gemm_cand_wmma(const float* __restrict__ S,
                                                      const float* __restrict__ Wp,
                                                      const float* __restrict__ bias,
                                                      const float* __restrict__ RU,
                                                      float* __restrict__ h1) {
  const int lane = threadIdx.x;
  const int wave = threadIdx.y;
  const int m0 = (blockIdx.x * 4 + wave) * 16;
  if (m0 >= N_NODES) return;
  const int n0 = blockIdx.y * 16;
  const int l15 = lane & 15;
  const int khi = (lane >> 4) << 1;
  const float* Srow = S + (size_t)(m0 + l15) * KPAD;
  v8f acc = {};
  for (int k = 0; k < KPAD; k += 4) {
    v2f a, b;
    a.x = Srow[k + khi];
    a.y = Srow[k + khi + 1];
    b.x = Wp[(size_t)(k + khi) * HID + n0 + l15];
    b.y = Wp[(size_t)(k + khi + 1) * HID + n0 + l15];
    acc = __builtin_amdgcn_wmma_f32_16x16x4_f32(false, a, false, b, (short)0, acc,
                                                false, false);
  }
  const float bv = bias[n0 + l15];
  const int rbase = m0 + ((lane >> 4) << 3);
  const int col = n0 + l15;
#pragma unroll
  for (int j = 0; j < 8; ++j) {
    const int row = rbase + j;
    float c = tanhf(acc[j] + bv);
    float u = RU[(size_t)row * RUW + HID + col];
    float h = h1[(size_t)row * HID + col];
    h1[(size_t)row * HID + col] = u * h + (1.0f - u) * c;
  }
}

// ------------------------------------------------------- layer-2 tiny GEMMs
// ru = sigmoid(S @ W2g + b2g), W2g:(65,2) row-major -> r2, u2 per node.
__global__ void l2_gate(const float* __restrict__ S, const float* __restrict__ Wg,
                        const float* __restrict__ bg, float* __restrict__ r2,
                        float* __restrict__ u2) {
  int n = blockIdx.x * 256 + threadIdx.x;
  if (n >= N_NODES) return;
  const float* s = S + (size_t)n * KPAD;
  float a0 = bg[0], a1 = bg[1];
  for (int k = 0; k < KDIM; ++k) {
    float sv = s[k];
    a0 += sv * Wg[k * 2 + 0];
    a1 += sv * Wg[k * 2 + 1];
  }
  r2[n] = sigmoidf_(a0);
  u2[n] = sigmoidf_(a1);
}

// c = tanh(S @ W2c + b2c); h2 = u*h2 + (1-u)*c; out_t[n] = h2.
__global__ void l2_cand(const float* __restrict__ S, const float* __restrict__ Wc,
                        const float* __restrict__ bc, const float* __restrict__ u2,
                        float* __restrict__ h2, float* __restrict__ out_t) {
  int n = blockIdx.x * 256 + threadIdx.x;
  if (n >= N_NODES) return;
  const float* s = S + (size_t)n * KPAD;
  float a = bc[0];
  for (int k = 0; k < KDIM; ++k) a += s[k] * Wc[k];
  float c = tanhf(a);
  float u = u2[n];
  float hn = u * h2[n] + (1.0f - u) * c;
  h2[n] = hn;
  out_t[n] = hn;
}

// ------------------------------------------------------------------- driver
extern "C" void kernel_launch(void* const* d_in, const int* in_sizes, int n_in,
                              void* d_out, int out_size, void* d_ws, size_t ws_size,
                              hipStream_t stream) {
  const float* features = (const float*)d_in[0];   // (T, N)
  const int*   index    = (const int*)d_in[1];     // (E, 2)
  const float* value    = (const float*)d_in[2];   // (E,)
  const float* W1_g = (const float*)d_in[3];
  const float* b1_g = (const float*)d_in[4];
  const float* W1_c = (const float*)d_in[5];
  const float* b1_c = (const float*)d_in[6];
  const float* W2b_g = (const float*)d_in[7];
  const float* b2b_g = (const float*)d_in[8];
  const float* W2b_c = (const float*)d_in[9];
  const float* b2b_c = (const float*)d_in[10];
  const float* W2u_g = (const float*)d_in[11];
  const float* b2u_g = (const float*)d_in[12];
  const float* W2u_c = (const float*)d_in[13];
  const float* b2u_c = (const float*)d_in[14];

  float* out_b = (float*)d_out;                    // (T, N)
  float* out_u = (float*)d_out + (size_t)T_STEPS * N_NODES;

  // ---- workspace carve-out (256B aligned slices of d_ws)
  char* wsp = (char*)d_ws;
  auto carve = [&](size_t bytes) {
    void* p = (void*)wsp;
    wsp += (bytes + 255) & ~((size_t)255);
    return p;
  };
  int*   rowptr  = (int*)carve((N_NODES + 1) * sizeof(int));
  int*   cursor  = (int*)carve(N_NODES * sizeof(int));  // also deg
  int*   csr_col = (int*)carve(N_EDGES * sizeof(int));
  float* csr_val = (float*)carve(N_EDGES * sizeof(float));
  float* xc      = (float*)carve((size_t)N_NODES * KPAD * sizeof(float));
  float* Sm      = (float*)carve((size_t)N_NODES * KPAD * sizeof(float));
  float* RU      = (float*)carve((size_t)N_NODES * RUW * sizeof(float));
  float* h1      = (float*)carve((size_t)N_NODES * HID * sizeof(float));
  float* h2b     = (float*)carve(N_NODES * sizeof(float));
  float* h2u     = (float*)carve(N_NODES * sizeof(float));
  float* r2      = (float*)carve(N_NODES * sizeof(float));
  float* u2      = (float*)carve(N_NODES * sizeof(float));
  float* W1g_pad = (float*)carve((size_t)KPAD * RUW * sizeof(float));
  float* W1c_pad = (float*)carve((size_t)KPAD * HID * sizeof(float));
  (void)in_sizes; (void)n_in; (void)out_size; (void)ws_size;

  const int B256_N  = (N_NODES + 255) / 256;
  const int B256_E  = (N_EDGES + 255) / 256;
  const int B256_XC = (N_NODES * KPAD + 255) / 256;
  const dim3 gemmBlk(32, 4);
  const int  mTiles = (N_NODES / 16 + 3) / 4;  // 50000/16 = 3125 tiles, 4 waves/blk

  // ---- one-time (per launch) CSR build + weight padding + state init
  fill_i32<<<B256_N, 256, 0, stream>>>(cursor, 0, N_NODES);
  hist_rows<<<B256_E, 256, 0, stream>>>(index, cursor);
  scan_deg<<<1, 1024, 0, stream>>>(cursor, rowptr, N_NODES);
  copy_i32<<<B256_N, 256, 0, stream>>>(rowptr, cursor, N_NODES);
  scatter_csr<<<B256_E, 256, 0, stream>>>(index, value, cursor, csr_col, csr_val);
  pad_weight<<<(KPAD * RUW + 255) / 256, 256, 0, stream>>>(W1_g, W1g_pad, RUW);
  pad_weight<<<(KPAD * HID + 255) / 256, 256, 0, stream>>>(W1_c, W1c_pad, HID);
  fill_f32<<<(N_NODES * HID + 255) / 256, 256, 0, stream>>>(h1, 0.0f, N_NODES * HID);
  fill_f32<<<B256_N, 256, 0, stream>>>(h2b, 0.0f, N_NODES);
  fill_f32<<<B256_N, 256, 0, stream>>>(h2u, 0.0f, N_NODES);

  // ---- time scan: layer-1 step, then both layer-2 steps (consume h1[t])
  for (int t = 0; t < T_STEPS; ++t) {
    const float* xt = features + (size_t)t * N_NODES;

    // Layer 1 gate
    build_xc_l1<<<B256_XC, 256, 0, stream>>>(xt, h1, RU, 0, xc);
    spmm_csr<<<(N_NODES + 7) / 8, 256, 0, stream>>>(rowptr, csr_col, csr_val, xc, Sm);
    gemm_gate_wmma<<<dim3(mTiles, RUW / 16), gemmBlk, 0, stream>>>(Sm, W1g_pad, b1_g, RU);
    // Layer 1 candidate + update (h1 <- new hidden)
    build_xc_l1<<<B256_XC, 256, 0, stream>>>(xt, h1, RU, 1, xc);
    spmm_csr<<<(N_NODES + 7) / 8, 256, 0, stream>>>(rowptr, csr_col, csr_val, xc, Sm);
    gemm_cand_wmma<<<dim3(mTiles, HID / 16), gemmBlk, 0, stream>>>(Sm, W1c_pad, b1_c, RU, h1);

    // Layer 2b
    build_xc_l2<<<B256_XC, 256, 0, stream>>>(h1, h2b, r2, 0, xc);
    spmm_csr<<<(N_NODES + 7) / 8, 256, 0, stream>>>(rowptr, csr_col, csr_val, xc, Sm);
    l2_gate<<<B256_N, 256, 0, stream>>>(Sm, W2b_g, b2b_g, r2, u2);
    build_xc_l2<<<B256_XC, 256, 0, stream>>>(h1, h2b, r2, 1, xc);
    spmm_csr<<<(N_NODES + 7) / 8, 256, 0, stream>>>(rowptr, csr_col, csr_val, xc, Sm);
    l2_cand<<<B256_N, 256, 0, stream>>>(Sm, W2b_c, b2b_c, u2, h2b,
                                        out_b + (size_t)t * N_NODES);

    // Layer 2u
    build_xc_l2<<<B256_XC, 256, 0, stream>>>(h1, h2u, r2, 0, xc);
    spmm_csr<<<(N_NODES + 7) / 8, 256, 0, stream>>>(rowptr, csr_col, csr_val, xc, Sm);
    l2_gate<<<B256_N, 256, 0, stream>>>(Sm, W2u_g, b2u_g, r2, u2);
    build_xc_l2<<<B256_XC, 256, 0, stream>>>(h1, h2u, r2, 1, xc);
    spmm_csr<<<(N_NODES + 7) / 8, 256, 0, stream>>>(rowptr, csr_col, csr_val, xc, Sm);
    l2_cand<<<B256_N, 256, 0, stream>>>(Sm, W2u_c, b2u_c, u2, h2u,
                                        out_u + (size_t)t * N_NODES);
  }
}